// CombinedLoss_72078141161702
// MI455X (gfx1250) — compile-verified
//
#include <hip/hip_runtime.h>
#include <hip/hip_bf16.h>

// ---------------------------------------------------------------------------
// CombinedLoss for MI455X (gfx1250, wave32).
//   B=4, F=16, N=1024, V=4
//   inputs: 0 refined_points [B,F,N,3] f32
//           1 gt_points      [B,F,N,3] f32
//           2 visibility     [B,F,N]   i32
//           3 projection     [V,B,3,4] f32
//           4 tracks_2d      [V,B,F,N,2] f32
//           5 corr           [V,V,B,N,N] f32   (268 MB -> bandwidth bound)
//   output: [total, rec, ident, att] f32
// ---------------------------------------------------------------------------

typedef __attribute__((ext_vector_type(2))) float v2f;
typedef __attribute__((ext_vector_type(8))) float v8f;

#define CL_B 4
#define CL_F 16
#define CL_N 1024
#define CL_V 4

// workspace layout (floats)
#define WS_REC   0       // 256 blocks * 8 floats
#define WS_IDENT 2048    // 256 floats
#define WS_ATT   2304    // 1024 floats

// ---------------- block reduction helpers (blockDim.x == 256) --------------
__device__ __forceinline__ float blkSum(float v, float* sm) {
  int t = threadIdx.x; sm[t] = v; __syncthreads();
  for (int s = 128; s > 0; s >>= 1) { if (t < s) sm[t] += sm[t + s]; __syncthreads(); }
  float r = sm[0]; __syncthreads(); return r;
}
__device__ __forceinline__ float blkMin(float v, float* sm) {
  int t = threadIdx.x; sm[t] = v; __syncthreads();
  for (int s = 128; s > 0; s >>= 1) { if (t < s) sm[t] = fminf(sm[t], sm[t + s]); __syncthreads(); }
  float r = sm[0]; __syncthreads(); return r;
}
__device__ __forceinline__ float blkMax(float v, float* sm) {
  int t = threadIdx.x; sm[t] = v; __syncthreads();
  for (int s = 128; s > 0; s >>= 1) { if (t < s) sm[t] = fmaxf(sm[t], sm[t + s]); __syncthreads(); }
  float r = sm[0]; __syncthreads(); return r;
}

// ---------------- kernel 1: reconstruction partials ------------------------
__global__ __launch_bounds__(256) void cl_rec_kernel(
    const float* __restrict__ pred, const float* __restrict__ gt,
    const int* __restrict__ vis, float* __restrict__ wsr) {
  __shared__ float sm[256];
  int p = blockIdx.x * 256 + threadIdx.x;            // 65536 points total
  bool m = vis[p] > 0;
  float g0 = gt[p * 3 + 0], g1 = gt[p * 3 + 1], g2 = gt[p * 3 + 2];
  float d0 = pred[p * 3 + 0] - g0;
  float d1 = pred[p * 3 + 1] - g1;
  float d2 = pred[p * 3 + 2] - g2;
  float se  = m ? (d0 * d0 + d1 * d1 + d2 * d2) : 0.0f;
  float cnt = m ? 1.0f : 0.0f;
  float mn0 = m ? g0 :  1e30f, mn1 = m ? g1 :  1e30f, mn2 = m ? g2 :  1e30f;
  float mx0 = m ? g0 : -1e30f, mx1 = m ? g1 : -1e30f, mx2 = m ? g2 : -1e30f;

  se  = blkSum(se, sm);  cnt = blkSum(cnt, sm);
  mn0 = blkMin(mn0, sm); mn1 = blkMin(mn1, sm); mn2 = blkMin(mn2, sm);
  mx0 = blkMax(mx0, sm); mx1 = blkMax(mx1, sm); mx2 = blkMax(mx2, sm);
  if (threadIdx.x == 0) {
    float* o = wsr + (size_t)blockIdx.x * 8;
    o[0] = se; o[1] = cnt; o[2] = mn0; o[3] = mn1; o[4] = mn2;
    o[5] = mx0; o[6] = mx1; o[7] = mx2;
  }
}

// ---------------- kernel 2: identity loss (WMMA projection) ----------------
// one block per (v,b,f); the 3x4 projection is done with V_WMMA_F32_16X16X4_F32:
//   A (16x4) = proj rows 0..2 padded with zeros, B (4x16) = 16 homogeneous pts,
//   D rows 0..2 = projected x,y,z for those 16 points.
__global__ __launch_bounds__(256) void cl_ident_kernel(
    const float* __restrict__ pred, const float* __restrict__ proj,
    const float* __restrict__ tracks, float* __restrict__ wsi) {
  __shared__ float sm[256];
  int vbf = blockIdx.x;                 // [V*B*F] = 256
  int f  = vbf % CL_F;
  int vb = vbf / CL_F;                  // v*B + b
  int b  = vb % CL_B;
  const float* tr = tracks + (size_t)(vb * CL_F + f) * CL_N * 2;
  const float* pp = pred   + (size_t)(b  * CL_F + f) * CL_N * 3;
  const float* P  = proj   + (size_t)vb * 12;
  int t = threadIdx.x;

  // ---- phase 1: per-(v,b,f) track min/max over valid points -> wh --------
  float mnx = 1e30f, mny = 1e30f, mxx = -1e30f, mxy = -1e30f;
  for (int k = 0; k < 4; ++k) {
    int n = t + 256 * k;
    float tx = tr[n * 2 + 0], ty = tr[n * 2 + 1];
    bool valid = (fabsf(tx) + fabsf(ty)) > 1e-6f;
    if (valid) {
      mnx = fminf(mnx, tx); mxx = fmaxf(mxx, tx);
      mny = fminf(mny, ty); mxy = fmaxf(mxy, ty);
    }
  }
  mnx = blkMin(mnx, sm); mny = blkMin(mny, sm);
  mxx = blkMax(mxx, sm); mxy = blkMax(mxy, sm);
  float whx = fmaxf(224.0f, mxx - mnx + 1e-6f);   // all-invalid => max(224,-inf)=224
  float why = fmaxf(224.0f, mxy - mny + 1e-6f);

  // ---- phase 2: WMMA projection + normalized squared error ---------------
  int lane = t & 31, wave = t >> 5;               // 8 waves of 32
  int mrow = lane & 15;                           // A: lane holds row M=lane&15
  int kb   = (lane >> 4) * 2;                     // VGPR0=K(kb), VGPR1=K(kb+1)
  v2f a;
  a.x = (mrow < 3) ? P[mrow * 4 + kb + 0] : 0.0f;
  a.y = (mrow < 3) ? P[mrow * 4 + kb + 1] : 0.0f;

  float acc = 0.0f;
  for (int g = 0; g < 8; ++g) {                   // 8 groups of 16 points/wave
    int npt = wave * 128 + g * 16 + mrow;         // B: lane holds column N=lane&15
    v2f bv;
    bv.x = (kb + 0 < 3) ? pp[npt * 3 + kb + 0] : 1.0f;   // homo coord kb
    bv.y = (kb + 1 < 3) ? pp[npt * 3 + kb + 1] : 1.0f;   // homo coord kb+1
    v8f c = {};
    c = __builtin_amdgcn_wmma_f32_16x16x4_f32(false, a, false, bv,
                                              (short)0, c, false, false);
    if (lane < 16) {                              // rows 0..2 live in lanes 0..15
      float inv = 1.0f / (c[2] + 1e-10f);
      float px = c[0] * inv, py = c[1] * inv;
      float tx = tr[npt * 2 + 0], ty = tr[npt * 2 + 1];
      float ex = (px - tx) / whx, ey = (py - ty) / why;
      acc += ex * ex + ey * ey;
    }
  }
  float s = blkSum(acc, sm);
  if (t == 0) wsi[blockIdx.x] = s;                // sum over 1024 points
}

// ---------------- kernel 3: attention BCE (streaming, WMMA reduction) ------
__device__ __forceinline__ float cl_bce(float x, float g) {
  // weights * (max(x,0) - x*g + log1p(exp(-|x|))),  weights = 1 + 2g
  float sp = fmaxf(x, 0.0f) + __logf(1.0f + __expf(-fabsf(x)));
  return fmaf(2.0f, g, 1.0f) * (sp - x * g);
}

__global__ __launch_bounds__(256) void cl_att_kernel(
    const float* __restrict__ corr, const int* __restrict__ vis,
    float* __restrict__ wsa) {
  __shared__ unsigned char visL[CL_N];
  __shared__ float sm[256];
  int chunk = blockIdx.x & 15;          // 16 chunks of 64 rows per matrix
  int mtx   = blockIdx.x >> 4;          // 0..63  (v1*V+v2)*B + b
  int b     = mtx & 3;
  int t = threadIdx.x;
  for (int n = t; n < CL_N; n += 256)
    visL[n] = (vis[b * CL_F * CL_N + n] > 0) ? 1 : 0;   // frame-0 visibility
  __syncthreads();

  const float4* base =
      reinterpret_cast<const float4*>(corr + (size_t)mtx * CL_N * CL_N +
                                      (size_t)chunk * 65536);
  v2f aOnes; aOnes.x = 1.0f; aOnes.y = 1.0f;   // ones A-matrix: D cols = sums
  v8f c = {};

  for (int kk = 0; kk < 8; ++kk) {
    float sx = 0.0f, sy = 0.0f;
    #pragma unroll
    for (int u = 0; u < 8; ++u) {
      int idx4 = (kk * 8 + u) * 256 + t;       // float4 index within chunk
      int e    = idx4 * 4;                     // element offset within chunk
      int i    = (chunk * 65536 + e) >> 10;    // row
      int j0   = e & 1023;                     // first of 4 columns
      __builtin_prefetch(base + idx4 + 2048, 0, 1);  // -> global_prefetch_b8
      float4 xv = base[idx4];
      float g0 = 0.0f, g1 = 0.0f, g2 = 0.0f, g3 = 0.0f;
      if ((unsigned)(i - j0 + 2) <= 7u) {      // band |i-j|<=2 can intersect
        bool vi = visL[i] != 0;
        #pragma unroll
        for (int q = 0; q < 4; ++q) {
          int j = j0 + q;
          int d = i - j; d = (d < 0) ? -d : d;
          float gb = (d == 0) ? 1.0f : (d == 1) ? 0.7f : (d == 2) ? 0.49f : 0.0f;
          float gv = (vi || visL[j] != 0) ? gb : 0.0f;
          if (q == 0) g0 = gv; else if (q == 1) g1 = gv;
          else if (q == 2) g2 = gv; else g3 = gv;
        }
      }
      sx += cl_bce(xv.x, g0) + cl_bce(xv.y, g1);
      sy += cl_bce(xv.z, g2) + cl_bce(xv.w, g3);
    }
    // f32 WMMA as a cross-lane reduction engine (f32 accumulate, no downcast):
    // every D entry becomes a column partial sum; 16x row replication divided
    // out in the finalize kernel.
    v2f bv; bv.x = sx; bv.y = sy;
    c = __builtin_amdgcn_wmma_f32_16x16x4_f32(false, aOnes, false, bv,
                                              (short)0, c, false, false);
  }
  float s = c[0] + c[1] + c[2] + c[3] + c[4] + c[5] + c[6] + c[7];
  s = blkSum(s, sm);                    // == 16 * (block element sum)
  if (t == 0) wsa[blockIdx.x] = s;
}

// ---------------- kernel 4: finalize ---------------------------------------
__global__ __launch_bounds__(256) void cl_final_kernel(
    const float* __restrict__ wsr, const float* __restrict__ wsi,
    const float* __restrict__ wsa, float* __restrict__ out) {
  __shared__ float sm[256];
  int t = threadIdx.x;
  const float* r = wsr + (size_t)t * 8;
  float se  = blkSum(r[0], sm);
  float cnt = blkSum(r[1], sm);
  float mn0 = blkMin(r[2], sm), mn1 = blkMin(r[3], sm), mn2 = blkMin(r[4], sm);
  float mx0 = blkMax(r[5], sm), mx1 = blkMax(r[6], sm), mx2 = blkMax(r[7], sm);
  float idsum = blkSum(wsi[t], sm);
  float atsum = blkSum(wsa[t] + wsa[t + 256] + wsa[t + 512] + wsa[t + 768], sm);
  if (t == 0) {
    float num   = 3.0f * cnt;
    float diff  = fmaxf(fmaxf(mx0 - mn0, mx1 - mn1), mx2 - mn2);
    float scale = (num > 0.0f) ? (diff + 1e-6f) : 1.0f;
    float rec   = (se / fmaxf(num, 1.0f)) / (scale * scale);
    float ident = idsum / (256.0f * 1024.0f);           // mean over (v,b,f) and n
    float att   = atsum / (16.0f * 67108864.0f);        // /16 WMMA replication
    out[0] = rec + 1.0f * ident + 0.5f * att;
    out[1] = rec;
    out[2] = ident;
    out[3] = att;
  }
}

// ---------------------------------------------------------------------------
extern "C" void kernel_launch(void* const* d_in, const int* in_sizes, int n_in,
                              void* d_out, int out_size, void* d_ws, size_t ws_size,
                              hipStream_t stream) {
  const float* pred   = (const float*)d_in[0];
  const float* gt     = (const float*)d_in[1];
  const int*   vis    = (const int*)d_in[2];
  const float* proj   = (const float*)d_in[3];
  const float* tracks = (const float*)d_in[4];
  const float* corr   = (const float*)d_in[5];
  float* out = (float*)d_out;
  float* ws  = (float*)d_ws;

  cl_rec_kernel  <<<256,  256, 0, stream>>>(pred, gt, vis, ws + WS_REC);
  cl_ident_kernel<<<256,  256, 0, stream>>>(pred, proj, tracks, ws + WS_IDENT);
  cl_att_kernel  <<<1024, 256, 0, stream>>>(corr, vis, ws + WS_ATT);
  cl_final_kernel<<<1,    256, 0, stream>>>(ws + WS_REC, ws + WS_IDENT,
                                            ws + WS_ATT, out);
}